// DeltaNetCell_65687229825727
// MI455X (gfx1250) — compile-verified
//
#include <hip/hip_runtime.h>

typedef __attribute__((ext_vector_type(2))) float v2f;
typedef __attribute__((ext_vector_type(4))) float v4f;
typedef __attribute__((ext_vector_type(8))) float v8f;

#define DDIM 128
#define ROWSTRIDE 132   // 128 + 4 pad: rows map to distinct LDS banks (132 mod 64 = 4)

#if defined(__gfx1250__) && __has_builtin(__builtin_amdgcn_global_load_async_to_lds_b128) && \
    __has_builtin(__builtin_amdgcn_s_wait_asynccnt)
#define USE_ASYNC_LDS 1
#else
#define USE_ASYNC_LDS 0
#endif

// Builtin expects: arg0 = int4* in global (AS1), arg1 = int4* in LDS (AS3), imm offset, imm cpol
typedef int v4i __attribute__((vector_size(16)));
typedef __attribute__((address_space(1))) v4i gbl_v4i;
typedef __attribute__((address_space(3))) v4i lds_v4i;

// One 256-thread block (8 wave32) per (b,h) matrix.
__global__ __launch_bounds__(256) void deltanet_cell_kernel(
    const float* __restrict__ q,
    const float* __restrict__ k,
    const float* __restrict__ v,
    const float* __restrict__ beta,
    const float* __restrict__ W_prev,
    float* __restrict__ o_out,
    float* __restrict__ W_out)
{
    extern __shared__ __align__(16) float smem[];
    float* lw   = smem;                       // 128 * 132 floats (padded W tile)
    float* lk   = smem + DDIM * ROWSTRIDE;    // 128
    float* lq   = lk + DDIM;                  // 128
    float* lv   = lq + DDIM;                  // 128
    float* lerr = lv + DDIM;                  // 128

    const int bh  = blockIdx.x;
    const int tid = threadIdx.x;
    const float beta_s = beta[bh];

    const float* Wg = W_prev + (size_t)bh * DDIM * DDIM;
    const float* kg = k + (size_t)bh * DDIM;
    const float* qg = q + (size_t)bh * DDIM;
    const float* vg = v + (size_t)bh * DDIM;

    // ---- Phase 1: stage k,q,v and W into LDS ----
    if (tid < DDIM) {
        lk[tid] = kg[tid];
        lq[tid] = qg[tid];
        lv[tid] = vg[tid];
    }
#if USE_ASYNC_LDS
    // Direct global->LDS DMA, no VGPR staging; tracked by ASYNCcnt.
    #pragma unroll
    for (int i = 0; i < 16; ++i) {
        int e   = (tid + i * 256) * 4;        // element index, 16B granule, same row
        int row = e >> 7;
        int col = e & 127;
        __builtin_amdgcn_global_load_async_to_lds_b128(
            (gbl_v4i*)(Wg + e),
            (lds_v4i*)(&lw[row * ROWSTRIDE + col]),
            0, 0);
    }
    __builtin_amdgcn_s_wait_asynccnt(0);
#else
    #pragma unroll
    for (int i = 0; i < 16; ++i) {
        int e   = (tid + i * 256) * 4;
        int row = e >> 7;
        int col = e & 127;
        v4f w = __builtin_nontemporal_load((const v4f*)(Wg + e));   // streamed once: TH=NT
        *(v4f*)(&lw[row * ROWSTRIDE + col]) = w;
    }
#endif
    __syncthreads();

    // ---- Phase 2: per-wave WMMA GEMV, 16 rows per wave ----
    const int lane = tid & 31;
    const int wv   = tid >> 5;      // wave 0..7
    const int n    = lane & 15;     // N / row-select within half
    const int hi   = lane >> 4;     // K-half select per WMMA A/B layout
    const int r0   = wv * 16;
    const int arow = r0 + n;        // A-matrix row M = lane%16

    v8f acc_k = {0.f,0.f,0.f,0.f,0.f,0.f,0.f,0.f};  // rows' dot with k
    v8f acc_q = {0.f,0.f,0.f,0.f,0.f,0.f,0.f,0.f};  // rows' dot with q

    #pragma unroll 4
    for (int base = 0; base < DDIM; base += 4) {
        // A layout (16x4 f32): lane L: M=L%16; VGPR0/1 = K = 2*(L>=16)+{0,1}
        v2f a  = *(const v2f*)(&lw[arow * ROWSTRIDE + base + 2 * hi]);
        // B (4x16) broadcast: every column n gets k[base+K]; same K striping as A
        v2f bk = *(const v2f*)(&lk[base + 2 * hi]);
        v2f bq = *(const v2f*)(&lq[base + 2 * hi]);
        acc_k = __builtin_amdgcn_wmma_f32_16x16x4_f32(
                    false, a, false, bk, (short)0, acc_k, false, false);
        acc_q = __builtin_amdgcn_wmma_f32_16x16x4_f32(
                    false, a, false, bq, (short)0, acc_q, false, false);
    }

    // kq = k . q (cheap redundant per-lane reduction; kernel is HBM-bound)
    float kq = 0.f;
    #pragma unroll 8
    for (int j = 0; j < DDIM; ++j) kq += lk[j] * lq[j];

    // C layout: VGPR t, lane L -> C[M = t + 8*(L>=16), N = L%16]; all N identical here.
    float err[8], ov[8];
    #pragma unroll
    for (int t = 0; t < 8; ++t) {
        int row = r0 + t + 8 * hi;
        float e = lv[row] - acc_k[t];            // err = v - W@k
        err[t] = e;
        ov[t]  = acc_q[t] + beta_s * kq * e;     // o = W@q + beta*(k.q)*err
    }
    if (n == 0) {        // lanes 0 and 16 publish their 8 rows each
        #pragma unroll
        for (int t = 0; t < 8; ++t) {
            int row = r0 + t + 8 * hi;
            lerr[row] = err[t];
            o_out[(size_t)bh * DDIM + row] = ov[t];
        }
    }
    __syncthreads();

    // ---- Phase 3: W_new = W + beta * err (outer) k, coalesced NT float4 stream ----
    float* Wn = W_out + (size_t)bh * DDIM * DDIM;
    #pragma unroll
    for (int i = 0; i < 16; ++i) {
        int e   = (tid + i * 256) * 4;
        int row = e >> 7;
        int col = e & 127;
        v4f w  = *(const v4f*)(&lw[row * ROWSTRIDE + col]);
        float be = beta_s * lerr[row];
        w.x += be * lk[col + 0];
        w.y += be * lk[col + 1];
        w.z += be * lk[col + 2];
        w.w += be * lk[col + 3];
        __builtin_nontemporal_store(w, (v4f*)(Wn + e));   // written once: TH=NT
    }
}

extern "C" void kernel_launch(void* const* d_in, const int* in_sizes, int n_in,
                              void* d_out, int out_size, void* d_ws, size_t ws_size,
                              hipStream_t stream) {
    const float* q     = (const float*)d_in[0];
    const float* k     = (const float*)d_in[1];
    const float* v     = (const float*)d_in[2];
    const float* beta  = (const float*)d_in[3];
    const float* W     = (const float*)d_in[4];

    const int BH = in_sizes[3];                 // B*H = 2048 (beta element count)
    float* o  = (float*)d_out;                  // o first in return order
    float* Wn = o + (size_t)BH * DDIM;          // then W_new

    const size_t lds_bytes = (size_t)(DDIM * ROWSTRIDE + 4 * DDIM) * sizeof(float); // ~68 KB
    deltanet_cell_kernel<<<BH, 256, lds_bytes, stream>>>(q, k, v, beta, W, o, Wn);
}